// BPNet_57836029608016
// MI455X (gfx1250) — compile-verified
//
#include <hip/hip_runtime.h>

typedef __attribute__((ext_vector_type(2))) float v2f;
typedef __attribute__((ext_vector_type(8))) float v8f;

#define NN 4096     // nodes
#define EE 16384    // edges
#define DD 13       // node feature dim
#define RR 128      // rank
#define PP 4        // NUM_PARAMS (edge types)
#define W1S 18      // stage1 LDS k-stride (16 padded + 2 anti-conflict)
#define W2S 130     // stage2 LDS k-stride (128 + 2 anti-conflict)

__global__ void zero_out_kernel(float* __restrict__ out, int n) {
  int i = blockIdx.x * blockDim.x + threadIdx.x;
  if (i < n) out[i] = 0.0f;
}

// ---------------------------------------------------------------------------
// Stage 1: t[slot][e][r] = relu(nodes[edges[e,slot]] @ W[type] + b[type])
// One wave handles one 16-edge tile; 16x16x4 f32 WMMA with per-row type
// masking folds the 4-way weight selection into 4 accumulating WMMAs.
// ---------------------------------------------------------------------------
__global__ __launch_bounds__(256) void bpnet_stage1_kernel(
    const float* __restrict__ nodes,
    const float* __restrict__ bp_params,   // [PP][DD][RR]
    const float* __restrict__ bp_bias,     // [PP][1][RR]
    const int*   __restrict__ edges,       // [EE][3]
    const int*   __restrict__ edge_types,  // [EE][3]
    float*       __restrict__ t_out)       // [3][EE][RR]
{
  __shared__ float Wt[PP * RR * W1S];   // [p][n][k] k zero-padded 13..17
  __shared__ float Bs[PP * RR];         // [p][n]

  const int tid = threadIdx.x;
  for (int idx = tid; idx < PP * RR * W1S; idx += 256) {
    int p = idx / (RR * W1S);
    int r = idx % (RR * W1S);
    int n = r / W1S;
    int k = r % W1S;
    Wt[idx] = (k < DD) ? bp_params[(p * DD + k) * RR + n] : 0.0f;
  }
  for (int idx = tid; idx < PP * RR; idx += 256) Bs[idx] = bp_bias[idx];
  __syncthreads();

  const int wave = tid >> 5;
  const int lane = tid & 31;
  const int row  = lane & 15;   // edge row within tile (A/C row, B/C col)
  const int kh   = lane >> 4;   // K half selector for A/B operands
  const int slot = blockIdx.y;
  const int tile = blockIdx.x * 8 + wave;
  const int e0   = tile * 16;
  const int e    = e0 + row;

  const int etype = edge_types[e * 3 + slot];
  const int nidx  = edges[e * 3 + slot];
  const int nbase = nidx * DD;

  // Per-lane A fragments for the 4 K-chunks (K padded 13 -> 16 with zeros)
  v2f av[4];
#pragma unroll
  for (int c = 0; c < 4; ++c) {
    int kk = 4 * c + 2 * kh;
    av[c].x = (kk     < DD) ? nodes[nbase + kk]     : 0.0f;
    av[c].y = (kk + 1 < DD) ? nodes[nbase + kk + 1] : 0.0f;
  }

  for (int nt = 0; nt < 8; ++nt) {
    const int ng = nt * 16 + row;  // global output column
    v8f acc = {0.f, 0.f, 0.f, 0.f, 0.f, 0.f, 0.f, 0.f};
#pragma unroll
    for (int c = 0; c < 4; ++c) {
      const int kk = 4 * c + 2 * kh;
#pragma unroll
      for (int p = 0; p < PP; ++p) {
        v2f b = *(const v2f*)&Wt[(p * RR + ng) * W1S + kk];
        v2f a;
        a.x = (etype == p) ? av[c].x : 0.0f;   // row-type mask => per-row W select
        a.y = (etype == p) ? av[c].y : 0.0f;
        acc = __builtin_amdgcn_wmma_f32_16x16x4_f32(
            false, a, false, b, (short)0, acc, false, false);
      }
    }
    // Epilogue: +bias[type], ReLU, store. C layout: VGPR v -> row v + 8*kh.
#pragma unroll
    for (int v = 0; v < 8; ++v) {
      int r2  = v + 8 * kh;
      int tr  = __shfl(etype, r2, 32);
      float x = acc[v] + Bs[tr * RR + ng];
      x = (x > 0.0f) ? x : 0.0f;
      t_out[(size_t)(slot * EE + (e0 + r2)) * RR + ng] = x;
    }
  }
}

// ---------------------------------------------------------------------------
// Stage 2: msg = (t[j0]*t[j1]) @ Who[slot][type] + b, scatter-add into out.
// K = 128 (32 chunks of 4), N = 13 padded to 16, same type-masking trick.
// ---------------------------------------------------------------------------
__global__ __launch_bounds__(256) void bpnet_stage2_kernel(
    const float* __restrict__ t_in,        // [3][EE][RR]
    const float* __restrict__ ho_params,   // [3][PP][RR][DD]
    const float* __restrict__ ho_bias,     // [3][PP][1][DD]
    const int*   __restrict__ edges,       // [EE][3]
    const int*   __restrict__ edge_types,  // [EE][3]
    float*       __restrict__ out)         // [NN][DD]
{
  __shared__ float Wt[PP * 16 * W2S];   // [p][n(16, pad>=13)][k(130, pad>=128)]
  __shared__ float Bs[PP * 16];

  const int slot = blockIdx.y;
  const int tid  = threadIdx.x;
  for (int idx = tid; idx < PP * 16 * W2S; idx += 256) {
    int p = idx / (16 * W2S);
    int r = idx % (16 * W2S);
    int n = r / W2S;
    int k = r % W2S;
    float v = 0.0f;
    if (n < DD && k < RR)
      v = ho_params[((size_t)(slot * PP + p) * RR + k) * DD + n];
    Wt[idx] = v;
  }
  for (int idx = tid; idx < PP * 16; idx += 256) {
    int p = idx / 16, n = idx % 16;
    Bs[idx] = (n < DD) ? ho_bias[(slot * PP + p) * DD + n] : 0.0f;
  }
  __syncthreads();

  const int wave = tid >> 5;
  const int lane = tid & 31;
  const int row  = lane & 15;
  const int kh   = lane >> 4;
  const int tile = blockIdx.x * 8 + wave;
  const int e0   = tile * 16;
  const int e    = e0 + row;

  const int etype = edge_types[e * 3 + slot];
  const int tgt   = edges[e * 3 + slot];
  const int j0    = (slot == 0) ? 1 : 0;
  const int j1    = (slot == 2) ? 1 : 2;

  const float* __restrict__ t0 = t_in + (size_t)(j0 * EE + e) * RR;
  const float* __restrict__ t1 = t_in + (size_t)(j1 * EE + e) * RR;

  v8f acc = {0.f, 0.f, 0.f, 0.f, 0.f, 0.f, 0.f, 0.f};
  for (int c = 0; c < 32; ++c) {
    const int kk = 4 * c + 2 * kh;
    v2f x0 = *(const v2f*)(t0 + kk);     // 8B aligned: row base & kk even
    v2f x1 = *(const v2f*)(t1 + kk);
    v2f a;
    a.x = x0.x * x1.x;                    // leave-one-out product on the fly
    a.y = x0.y * x1.y;
#pragma unroll
    for (int p = 0; p < PP; ++p) {
      v2f b = *(const v2f*)&Wt[(p * 16 + row) * W2S + kk];
      v2f ap;
      ap.x = (etype == p) ? a.x : 0.0f;
      ap.y = (etype == p) ? a.y : 0.0f;
      acc = __builtin_amdgcn_wmma_f32_16x16x4_f32(
          false, ap, false, b, (short)0, acc, false, false);
    }
  }

  // Epilogue: +bias[type], scatter-add to target node rows (cols 0..12 valid)
#pragma unroll
  for (int v = 0; v < 8; ++v) {
    int r2 = v + 8 * kh;
    int tr = __shfl(etype, r2, 32);
    int tg = __shfl(tgt,   r2, 32);
    if (row < DD) {
      float val = acc[v] + Bs[tr * 16 + row];
      atomicAdd(&out[tg * DD + row], val);
    }
  }
}

extern "C" void kernel_launch(void* const* d_in, const int* in_sizes, int n_in,
                              void* d_out, int out_size, void* d_ws, size_t ws_size,
                              hipStream_t stream) {
  const float* nodes      = (const float*)d_in[0];
  const float* bp_params  = (const float*)d_in[1];
  const float* bp_bias    = (const float*)d_in[2];
  const float* ho_params  = (const float*)d_in[3];
  const float* ho_bias    = (const float*)d_in[4];
  const int*   edges      = (const int*)d_in[5];
  const int*   edge_types = (const int*)d_in[6];
  // d_in[7]=atoms, d_in[8]=atom_edges unused in BT mode
  float* out  = (float*)d_out;
  float* t_ws = (float*)d_ws;   // [3][EE][RR] f32 = 25.2 MB

  // Zero output every call: stage2 uses global f32 atomics, and the harness
  // replays this launch without re-poisoning buffers.
  zero_out_kernel<<<(NN * DD + 255) / 256, 256, 0, stream>>>(out, NN * DD);

  // 1024 tiles per slot, 8 waves per 256-thread block -> grid (128, 3)
  bpnet_stage1_kernel<<<dim3(EE / 16 / 8, 3), 256, 0, stream>>>(
      nodes, bp_params, bp_bias, edges, edge_types, t_ws);
  bpnet_stage2_kernel<<<dim3(EE / 16 / 8, 3), 256, 0, stream>>>(
      t_ws, ho_params, ho_bias, edges, edge_types, out);
}